// DGCNN_Grouper_43997644980255
// MI455X (gfx1250) — compile-verified
//
#include <hip/hip_runtime.h>
#include <hip/hip_bf16.h>
#include <stdint.h>

typedef __attribute__((ext_vector_type(2))) float v2f;
typedef __attribute__((ext_vector_type(8))) float v8f;

#define KNN_K 16
#define FPS_T 256
#define KNN_WAVES 4  // waves per block in knn_wmma_kernel

// ---------------------------------------------------------------------------
// feat = Wt @ pc + bt   (Wt: [8,3], pc: [B,3,N] -> feat: [B,8,N])
// ---------------------------------------------------------------------------
__global__ void feat_init_kernel(const float* __restrict__ pc,
                                 const float* __restrict__ Wt,
                                 const float* __restrict__ bt,
                                 float* __restrict__ feat, int B, int N) {
  int i = blockIdx.x * blockDim.x + threadIdx.x;
  int total = B * 8 * N;
  if (i >= total) return;
  int n = i % N;
  int o = (i / N) % 8;
  int b = i / (8 * N);
  const float* p = pc + (size_t)b * 3 * N;
  feat[i] = Wt[o * 3 + 0] * p[n] + Wt[o * 3 + 1] * p[N + n] +
            Wt[o * 3 + 2] * p[2 * N + n] + bt[o];
}

// ---------------------------------------------------------------------------
// sq[b,n] = ||coords[b,:,n]||^2
// ---------------------------------------------------------------------------
__global__ void sqnorm_kernel(const float* __restrict__ c,
                              float* __restrict__ sq, int B, int N) {
  int i = blockIdx.x * blockDim.x + threadIdx.x;
  if (i >= B * N) return;
  int n = i % N, b = i / N;
  const float* cb = c + (size_t)b * 3 * N;
  float x = cb[n], y = cb[N + n], z = cb[2 * N + n];
  sq[i] = x * x + y * y + z * z;
}

// ---------------------------------------------------------------------------
// kNN via WMMA distance tiles.
// d2[m][n] = q2[m] + k2[n] - 2*q[m]·k[n]  (matches the reference formula)
// encoded as one V_WMMA_F32_16X16X4_F32 per 16x16 tile:
//   A[m][0..3] = (-2qx, -2qy, -2qz, 1),  B[0..3][n] = (kx, ky, kz, ||k||^2),
//   C init    = q2[m] (row constant).
// One wave owns 32 queries (two 16-row M tiles). Per 16-key tile: 2 WMMAs,
// spill both D tiles to this wave's private LDS slice, then each lane runs a
// register-resident top-16 insertion on its own query row. No barriers
// needed: each wave owns its LDS slice and per-wave LDS ops are in-order.
// Grid: x = Q/(32*KNN_WAVES), z = B; block = 32*KNN_WAVES (EXEC always full).
// ---------------------------------------------------------------------------
__global__ void knn_wmma_kernel(const float* __restrict__ qc,   // [B][3][Q]
                                const float* __restrict__ kc,   // [B][3][N]
                                const float* __restrict__ qsq,  // [B][Q]
                                const float* __restrict__ ksq,  // [B][N]
                                int* __restrict__ idx,          // [B][Q][16]
                                int Q, int N) {
  __shared__ float tile[KNN_WAVES][32][17];  // +1 pad: bank-conflict-free rows
  const int b = blockIdx.z;
  const int lane = threadIdx.x & 31;
  const int wv = threadIdx.x >> 5;
  const int q0 = (blockIdx.x * KNN_WAVES + wv) * 32;
  const bool hi = (lane >= 16);
  const int col = lane & 15;

  const float* qcb = qc + (size_t)b * 3 * Q;
  const float* kcb = kc + (size_t)b * 3 * N;
  const float* qsqb = qsq + (size_t)b * Q;
  const float* ksqb = ksq + (size_t)b * N;

  // A matrices (wave-invariant): tile0 rows q0..q0+15, tile1 rows q0+16..q0+31.
  const int m0 = q0 + col;
  const int m1 = q0 + 16 + col;
  v2f a0, a1;
  if (!hi) {  // K = 0,1 -> -2*qx, -2*qy
    a0.x = -2.0f * qcb[m0];       a0.y = -2.0f * qcb[Q + m0];
    a1.x = -2.0f * qcb[m1];       a1.y = -2.0f * qcb[Q + m1];
  } else {    // K = 2,3 -> -2*qz, 1
    a0.x = -2.0f * qcb[2 * Q + m0]; a0.y = 1.0f;
    a1.x = -2.0f * qcb[2 * Q + m1]; a1.y = 1.0f;
  }

  // C init row constants (hoisted): q2 of each D row.
  v8f cin0, cin1;
#pragma unroll
  for (int r = 0; r < 8; ++r) {
    cin0[r] = qsqb[q0 + r + (hi ? 8 : 0)];
    cin1[r] = qsqb[q0 + 16 + r + (hi ? 8 : 0)];
  }

  float bd[KNN_K];
  int bi[KNN_K];
#pragma unroll
  for (int r = 0; r < KNN_K; ++r) { bd[r] = 3.4e38f; bi[r] = 0; }

  const int ntiles = N / 16;
  for (int nt = 0; nt < ntiles; ++nt) {
    const int n = nt * 16 + col;
    __builtin_prefetch(kcb + n + 128, 0, 0);  // global_prefetch_b8
    v2f bm;
    if (!hi) { bm.x = kcb[n];         bm.y = kcb[N + n]; }
    else     { bm.x = kcb[2 * N + n]; bm.y = ksqb[n]; }

    v8f acc0 = __builtin_amdgcn_wmma_f32_16x16x4_f32(
        false, a0, false, bm, (short)0, cin0, false, false);
    v8f acc1 = __builtin_amdgcn_wmma_f32_16x16x4_f32(
        false, a1, false, bm, (short)0, cin1, false, false);

    // Spill D tiles to this wave's LDS slice (row = query-in-wave, col = key).
#pragma unroll
    for (int r = 0; r < 8; ++r) {
      tile[wv][r + (hi ? 8 : 0)][col] = acc0[r];
      tile[wv][16 + r + (hi ? 8 : 0)][col] = acc1[r];
    }

    // Each lane selects over its own query row (ascending key index).
#pragma unroll
    for (int c = 0; c < 16; ++c) {
      float d = tile[wv][lane][c];
      int nn = nt * 16 + c;
      if (d < bd[KNN_K - 1]) {
        bd[KNN_K - 1] = d;
        bi[KNN_K - 1] = nn;
#pragma unroll
        for (int j = KNN_K - 1; j > 0; --j) {
          if (bd[j] < bd[j - 1]) {
            float td = bd[j]; bd[j] = bd[j - 1]; bd[j - 1] = td;
            int ti = bi[j]; bi[j] = bi[j - 1]; bi[j - 1] = ti;
          }
        }
      }
    }
  }

  int* op = idx + ((size_t)b * Q + q0 + lane) * KNN_K;
#pragma unroll
  for (int r = 0; r < KNN_K; ++r) op[r] = bi[r];
}

// ---------------------------------------------------------------------------
// bias[b,o,q] = sum_c (W[o][C+c] - W[o][c]) * qf[b,c,q]
// (folds the [neigh - q, q] concat: h = W1p@neigh + (W2p-W1p)@q)
// ---------------------------------------------------------------------------
__global__ void edge_bias_kernel(const float* __restrict__ W,
                                 const float* __restrict__ qf,
                                 float* __restrict__ bias,
                                 int B, int C, int Cout, int Q) {
  int i = blockIdx.x * blockDim.x + threadIdx.x;
  int total = B * Cout * Q;
  if (i >= total) return;
  int q = i % Q;
  int o = (i / Q) % Cout;
  int b = i / (Q * Cout);
  const float* Wr = W + (size_t)o * 2 * C;
  const float* qfb = qf + (size_t)b * C * Q;
  float s = 0.0f;
  for (int c = 0; c < C; ++c) s += (Wr[C + c] - Wr[c]) * qfb[(size_t)c * Q + q];
  bias[i] = s;
}

__global__ void zero_kernel(float* __restrict__ p, int n) {
  int i = blockIdx.x * blockDim.x + threadIdx.x;
  if (i < n) p[i] = 0.0f;
}

// ---------------------------------------------------------------------------
// Edge GEMM via V_WMMA_F32_16X16X4_F32.
// One wave computes a 16(out-ch) x 16(neighbor) tile for one query q:
//   D = sum_{c step 4} A(16x4 = W1p[m0..m0+15, c..c+3]) x
//                      B(4x16 = kf[c..c+3, idx[q, 0..15]])  + bias column.
// Also accumulates GroupNorm sum / sumsq per (batch, group) via LDS atomics.
// h layout: [B, Cout, Q, 16]; stats layout: [B, 4, 2] (sum, sumsq).
// Grid: x = Q*(Cout/16)/4, z = B; block = 128 (4 waves, EXEC always full).
// ---------------------------------------------------------------------------
__global__ void edge_gemm_wmma_kernel(const float* __restrict__ W,   // [Cout][2C]
                                      const float* __restrict__ kf,  // [B][C][N]
                                      const int* __restrict__ idx,   // [B][Q][16]
                                      const float* __restrict__ bias,// [B][Cout][Q]
                                      float* __restrict__ h,         // [B][Cout][Q][16]
                                      float* __restrict__ stats,     // [B][4][2]
                                      int C, int Cout, int Q, int N) {
  __shared__ float s_sum[4];
  __shared__ float s_sq[4];
  if (threadIdx.x < 4) { s_sum[threadIdx.x] = 0.0f; s_sq[threadIdx.x] = 0.0f; }
  __syncthreads();

  const int b = blockIdx.z;
  const int lane = threadIdx.x & 31;
  const int wave = threadIdx.x >> 5;
  const int mtiles = Cout >> 4;
  const int tile = blockIdx.x * 4 + wave;  // grid sized exactly: tile < Q*mtiles
  const int q = tile / mtiles;
  const int m0 = (tile % mtiles) << 4;

  const bool hi = (lane >= 16);
  const int col = lane & 15;                 // neighbor slot / D column
  const int nidx = idx[((size_t)b * Q + q) * KNN_K + col];
  const float* kfb = kf + (size_t)b * C * N;
  const int mrow = m0 + (lane & 15);         // A-matrix row for this lane

  // C accumulator initialized with the per-query bias column.
  const float* biasb = bias + ((size_t)b * Cout) * Q + q;
  v8f acc;
#pragma unroll
  for (int r = 0; r < 8; ++r)
    acc[r] = biasb[(size_t)(m0 + r + (hi ? 8 : 0)) * Q];

  for (int c = 0; c < C; c += 4) {
    const int kc4 = c + (hi ? 2 : 0);
    v2f a;
    a.x = W[(size_t)mrow * (2 * C) + kc4];
    a.y = W[(size_t)mrow * (2 * C) + kc4 + 1];
    v2f bm;
    bm.x = kfb[(size_t)kc4 * N + nidx];
    bm.y = kfb[(size_t)(kc4 + 1) * N + nidx];
    acc = __builtin_amdgcn_wmma_f32_16x16x4_f32(
        /*neg_a=*/false, a, /*neg_b=*/false, bm,
        /*c_mod=*/(short)0, acc, /*reuse_a=*/false, /*reuse_b=*/false);
  }

  // Store D tile + accumulate GroupNorm statistics.
  float* hb = h + (((size_t)b * Cout) * Q + q) * 16;
  const int gsz = Cout >> 2;  // channels per group (4 groups)
#pragma unroll
  for (int r = 0; r < 8; ++r) {
    const int m = m0 + r + (hi ? 8 : 0);
    const float v = acc[r];
    hb[(size_t)m * Q * 16 + col] = v;
    const int g = m / gsz;
    atomicAdd(&s_sum[g], v);       // LDS ds_add_f32
    atomicAdd(&s_sq[g], v * v);
  }
  __syncthreads();
  if (threadIdx.x < 4) {
    atomicAdd(&stats[(b * 4 + threadIdx.x) * 2 + 0], s_sum[threadIdx.x]);
    atomicAdd(&stats[(b * 4 + threadIdx.x) * 2 + 1], s_sq[threadIdx.x]);
  }
}

// ---------------------------------------------------------------------------
// GroupNorm (4 groups) + LeakyReLU(0.2) + max over the 16 neighbors.
// ---------------------------------------------------------------------------
__global__ void gn_max_kernel(const float* __restrict__ h,
                              const float* __restrict__ stats,
                              const float* __restrict__ gamma,
                              const float* __restrict__ beta,
                              float* __restrict__ out,
                              int B, int Cout, int Q) {
  int i = blockIdx.x * blockDim.x + threadIdx.x;
  int total = B * Cout * Q;
  if (i >= total) return;
  int q = i % Q;
  int o = (i / Q) % Cout;
  int b = i / (Q * Cout);
  const int gsz = Cout >> 2;
  const int g = o / gsz;
  const float cnt = (float)gsz * (float)Q * 16.0f;
  const float su = stats[(b * 4 + g) * 2 + 0];
  const float sq = stats[(b * 4 + g) * 2 + 1];
  const float mu = su / cnt;
  const float var = sq / cnt - mu * mu;
  const float inv = rsqrtf(var + 1e-5f);
  const float ga = gamma[o], be = beta[o];
  const float* hp = h + (((size_t)b * Cout + o) * Q + q) * 16;
  float m = -3.4e38f;
#pragma unroll
  for (int k = 0; k < 16; ++k) {
    float x = (hp[k] - mu) * inv * ga + be;
    x = (x >= 0.0f) ? x : 0.2f * x;
    m = fmaxf(m, x);
  }
  out[((size_t)b * Cout + o) * Q + q] = m;
}

// ---------------------------------------------------------------------------
// Farthest point sampling. One block per batch; sequential scan with LDS
// argmax reduction (tie -> lowest index, matching jnp.argmax).
// ---------------------------------------------------------------------------
__global__ void fps_kernel(const float* __restrict__ coords,  // [B][3][N]
                           int N, int n_out, int* __restrict__ idx_out) {
  const int b = blockIdx.x;
  const int t = threadIdx.x;
  const float* cb = coords + (size_t)b * 3 * N;
  __shared__ float rd[FPS_T];
  __shared__ int ri[FPS_T];
  float mind[16];
#pragma unroll
  for (int p = 0; p < 16; ++p) mind[p] = 1e10f;
  int cur = 0;
  if (t == 0) idx_out[b * n_out] = 0;
  const int P = (N + FPS_T - 1) / FPS_T;
  for (int it = 1; it < n_out; ++it) {
    const float px = cb[cur], py = cb[N + cur], pz = cb[2 * N + cur];
    float bdv = -1.0f;
    int biv = 0;
#pragma unroll
    for (int p = 0; p < 16; ++p) {
      int n = t + p * FPS_T;
      if (p < P && n < N) {
        float dx = cb[n] - px, dy = cb[N + n] - py, dz = cb[2 * N + n] - pz;
        float d = dx * dx + dy * dy + dz * dz;
        float m = fminf(mind[p], d);
        mind[p] = m;
        if (m > bdv) { bdv = m; biv = n; }
      }
    }
    rd[t] = bdv;
    ri[t] = biv;
    __syncthreads();
    for (int s = FPS_T / 2; s > 0; s >>= 1) {
      if (t < s) {
        float od = rd[t + s];
        int oi = ri[t + s];
        if (od > rd[t] || (od == rd[t] && oi < ri[t])) { rd[t] = od; ri[t] = oi; }
      }
      __syncthreads();
    }
    cur = ri[0];
    if (t == 0) idx_out[b * n_out + it] = cur;
    __syncthreads();
  }
}

// ---------------------------------------------------------------------------
// Column gather: dst[b,c,j] = src[b,c, fidx[b,j]]
// ---------------------------------------------------------------------------
__global__ void gather_kernel(const float* __restrict__ src,
                              const int* __restrict__ fidx,
                              float* __restrict__ dst,
                              int B, int C, int Nsrc, int Nout) {
  int i = blockIdx.x * blockDim.x + threadIdx.x;
  int total = B * C * Nout;
  if (i >= total) return;
  int j = i % Nout;
  int c = (i / Nout) % C;
  int b = i / (C * Nout);
  dst[i] = src[((size_t)b * C + c) * Nsrc + fidx[b * Nout + j]];
}

// ---------------------------------------------------------------------------
// Host-side pipeline
// ---------------------------------------------------------------------------
extern "C" void kernel_launch(void* const* d_in, const int* in_sizes, int n_in,
                              void* d_out, int out_size, void* d_ws, size_t ws_size,
                              hipStream_t stream) {
  const int B = 8, N0 = 4096, N1 = 1024, N2 = 256;
  const float* pc = (const float*)d_in[0];
  const float* Wt = (const float*)d_in[1];
  const float* bt = (const float*)d_in[2];
  const float* W1 = (const float*)d_in[3];
  const float* g1 = (const float*)d_in[4];
  const float* b1 = (const float*)d_in[5];
  const float* W2 = (const float*)d_in[6];
  const float* g2 = (const float*)d_in[7];
  const float* b2 = (const float*)d_in[8];
  const float* W3 = (const float*)d_in[9];
  const float* g3 = (const float*)d_in[10];
  const float* b3 = (const float*)d_in[11];
  const float* W4 = (const float*)d_in[12];
  const float* g4 = (const float*)d_in[13];
  const float* b4 = (const float*)d_in[14];

  // Workspace carve-out
  char* w = (char*)d_ws;
  float* feat0 = (float*)w;           w += (size_t)B * 8 * N0 * 4;
  float* kf1   = (float*)w;           w += (size_t)B * 32 * N0 * 4;
  float* qc1   = (float*)w;           w += (size_t)B * 3 * N1 * 4;
  float* qf1   = (float*)w;           w += (size_t)B * 32 * N1 * 4;
  float* qf2   = (float*)w;           w += (size_t)B * 64 * N1 * 4;
  float* kf3   = (float*)w;           w += (size_t)B * 64 * N1 * 4;
  float* qf3d  = (float*)w;           w += (size_t)B * 64 * N2 * 4;
  float* sqbuf = (float*)w;           w += (size_t)B * N0 * 4;            // ||.||^2
  float* bias  = (float*)w;           w += (size_t)B * 32 * N0 * 4;      // max bias
  float* hbuf  = (float*)w;           w += (size_t)B * 32 * N0 * 16 * 4; // max h
  float* stats = (float*)w;           w += 64 * 4;
  int* idxknn  = (int*)w;             w += (size_t)B * N0 * KNN_K * 4;
  int* idxfps  = (int*)w;             w += (size_t)B * N1 * 4;

  float* out_qc = (float*)d_out;                       // [8,3,256]
  float* out_qf = (float*)d_out + (size_t)B * 3 * N2;  // [8,128,256]

  const int TB = 256;
  auto blocks = [](int n, int t) { return (n + t - 1) / t; };

  // ---- feat = Wt @ pc + bt ------------------------------------------------
  feat_init_kernel<<<blocks(B * 8 * N0, TB), TB, 0, stream>>>(pc, Wt, bt, feat0, B, N0);

  // ---- edge_conv 1: (pc, feat0) -> kf1 [8,32,4096], C=8, Cout=32 ---------
  sqnorm_kernel<<<blocks(B * N0, TB), TB, 0, stream>>>(pc, sqbuf, B, N0);
  knn_wmma_kernel<<<dim3(N0 / (32 * KNN_WAVES), 1, B), 32 * KNN_WAVES, 0, stream>>>(
      pc, pc, sqbuf, sqbuf, idxknn, N0, N0);
  edge_bias_kernel<<<blocks(B * 32 * N0, TB), TB, 0, stream>>>(W1, feat0, bias, B, 8, 32, N0);
  zero_kernel<<<1, 64, 0, stream>>>(stats, 64);
  edge_gemm_wmma_kernel<<<dim3(N0 * 2 / 4, 1, B), 128, 0, stream>>>(
      W1, feat0, idxknn, bias, hbuf, stats, 8, 32, N0, N0);
  gn_max_kernel<<<blocks(B * 32 * N0, TB), TB, 0, stream>>>(hbuf, stats, g1, b1, kf1, B, 32, N0);

  // ---- FPS 4096 -> 1024, gather coords + feats ---------------------------
  fps_kernel<<<B, FPS_T, 0, stream>>>(pc, N0, N1, idxfps);
  gather_kernel<<<blocks(B * 3 * N1, TB), TB, 0, stream>>>(pc, idxfps, qc1, B, 3, N0, N1);
  gather_kernel<<<blocks(B * 32 * N1, TB), TB, 0, stream>>>(kf1, idxfps, qf1, B, 32, N0, N1);

  // ---- edge_conv 2: (qc1, qf1) -> qf2 [8,64,1024], C=32, Cout=64 ---------
  sqnorm_kernel<<<blocks(B * N1, TB), TB, 0, stream>>>(qc1, sqbuf, B, N1);
  knn_wmma_kernel<<<dim3(N1 / (32 * KNN_WAVES), 1, B), 32 * KNN_WAVES, 0, stream>>>(
      qc1, qc1, sqbuf, sqbuf, idxknn, N1, N1);
  edge_bias_kernel<<<blocks(B * 64 * N1, TB), TB, 0, stream>>>(W2, qf1, bias, B, 32, 64, N1);
  zero_kernel<<<1, 64, 0, stream>>>(stats, 64);
  edge_gemm_wmma_kernel<<<dim3(N1 * 4 / 4, 1, B), 128, 0, stream>>>(
      W2, qf1, idxknn, bias, hbuf, stats, 32, 64, N1, N1);
  gn_max_kernel<<<blocks(B * 64 * N1, TB), TB, 0, stream>>>(hbuf, stats, g2, b2, qf2, B, 64, N1);

  // ---- edge_conv 3: (qc1, qf2) -> kf3 [8,64,1024], C=64, Cout=64 ---------
  // coords unchanged -> kNN indices identical to edge_conv 2; reuse idxknn.
  edge_bias_kernel<<<blocks(B * 64 * N1, TB), TB, 0, stream>>>(W3, qf2, bias, B, 64, 64, N1);
  zero_kernel<<<1, 64, 0, stream>>>(stats, 64);
  edge_gemm_wmma_kernel<<<dim3(N1 * 4 / 4, 1, B), 128, 0, stream>>>(
      W3, qf2, idxknn, bias, hbuf, stats, 64, 64, N1, N1);
  gn_max_kernel<<<blocks(B * 64 * N1, TB), TB, 0, stream>>>(hbuf, stats, g3, b3, kf3, B, 64, N1);

  // ---- FPS 1024 -> 256, gather coords (direct to d_out) + feats ----------
  fps_kernel<<<B, FPS_T, 0, stream>>>(qc1, N1, N2, idxfps);
  gather_kernel<<<blocks(B * 3 * N2, TB), TB, 0, stream>>>(qc1, idxfps, out_qc, B, 3, N1, N2);
  gather_kernel<<<blocks(B * 64 * N2, TB), TB, 0, stream>>>(kf3, idxfps, qf3d, B, 64, N1, N2);

  // ---- edge_conv 4: (out_qc, qf3d) -> out_qf [8,128,256], C=64, Cout=128 -
  sqnorm_kernel<<<blocks(B * N2, TB), TB, 0, stream>>>(out_qc, sqbuf, B, N2);
  knn_wmma_kernel<<<dim3(N2 / (32 * KNN_WAVES), 1, B), 32 * KNN_WAVES, 0, stream>>>(
      out_qc, out_qc, sqbuf, sqbuf, idxknn, N2, N2);
  edge_bias_kernel<<<blocks(B * 128 * N2, TB), TB, 0, stream>>>(W4, qf3d, bias, B, 64, 128, N2);
  zero_kernel<<<1, 64, 0, stream>>>(stats, 64);
  edge_gemm_wmma_kernel<<<dim3(N2 * 8 / 4, 1, B), 128, 0, stream>>>(
      W4, qf3d, idxknn, bias, hbuf, stats, 64, 128, N2, N2);
  gn_max_kernel<<<blocks(B * 128 * N2, TB), TB, 0, stream>>>(hbuf, stats, g4, b4, out_qf, B, 128, N2);
}